// GridEmaInterpolate_39685497815537
// MI455X (gfx1250) — compile-verified
//
#include <hip/hip_runtime.h>
#include <hip/hip_bf16.h>

typedef __attribute__((ext_vector_type(2))) float v2f;
typedef __attribute__((ext_vector_type(8))) float v8f;

#define C_CH 32   // channels (== wave32 width)
#define F_FR 16   // frames   (== WMMA tile dim)
#define WAVES 8   // waves (segments) per block

// fast softplus using hardware transcendentals (v_exp_f32 / v_log_f32):
// softplus(x) = max(x,0) + log(1 + exp(-|x|))   (argument of log in [1,2] -> well conditioned)
__device__ __forceinline__ float softplus_fast(float x) {
    return fmaxf(x, 0.0f) + __logf(1.0f + __expf(-fabsf(x)));
}

__device__ __forceinline__ int lower_bound_i32(const int* __restrict__ ids, int n, int key) {
    int lo = 0, hi = n;
    while (lo < hi) {
        int mid = (lo + hi) >> 1;
        if (ids[mid] < key) lo = mid + 1; else hi = mid;
    }
    return lo;
}

__global__ __launch_bounds__(C_CH * WAVES)
void GridEmaInterpolate_kernel(const float* __restrict__ features,
                               const float* __restrict__ times,
                               const float* __restrict__ decay,
                               const int*   __restrict__ seg_ids,
                               const float* __restrict__ t_start,
                               const float* __restrict__ t_stop,
                               float* __restrict__ out,
                               int E, int G, int numSeg) {
    __shared__ float Pl[WAVES][F_FR * C_CH];   // per-wave bucket sums P[fb][c]

    const int wave = threadIdx.x >> 5;
    const int lane = threadIdx.x & 31;
    int seg = blockIdx.x * WAVES + wave;
    if (seg >= numSeg) seg = numSeg - 1;       // clamp (uniform per wave; duplicate writes benign)

    const int   b      = seg / G;
    const float ts     = t_start[b];
    const float dt     = (t_stop[b] - ts) * (1.0f / (float)F_FR);
    const float inv_dt = 1.0f / dt;

    // ---- Phase 1: per-segment bucket accumulation (lane == channel) ----
    const float rl = softplus_fast(decay[lane]);   // rate for this lane's channel
    float* Pw = &Pl[wave][0];
#pragma unroll
    for (int fb = 0; fb < F_FR; ++fb) Pw[fb * C_CH + lane] = 0.0f;

    // uniform binary search for this segment's event range (segment_ids sorted)
    const int e0 = lower_bound_i32(seg_ids, E, seg);
    const int e1 = lower_bound_i32(seg_ids, E, seg + 1);

    for (int e = e0; e < e1; ++e) {
        const float t = times[e];                          // uniform broadcast load
        int fmin = (int)ceilf((t - ts) * inv_dt) - 1;      // first valid frame
        if (fmin < 0) fmin = 0;
        if (fmin >= F_FR) continue;                        // uniform: event after window
        // factored EMA weight: exp(t*r); frame factor exp(-frame_t*r) applied at the end
        const float a = __expf(t * rl) * features[(size_t)e * C_CH + lane];
        Pw[fmin * C_CH + lane] += a;                       // lane-private column: no conflicts
    }
    __syncthreads();

    // ---- Phase 2: inclusive frame-cumsum S = L(16x16 lower-tri ones) x P(16x32) ----
    // via V_WMMA_F32_16X16X4_F32, 4 K-chunks x 2 N-halves.
    const int hi16 = lane >> 4;     // 0: lanes 0-15, 1: lanes 16-31
    const int lrow = lane & 15;
    float* outseg = out + (size_t)seg * (F_FR * C_CH);     // out[seg][f][c]

#pragma unroll
    for (int h = 0; h < 2; ++h) {                          // channel half: c in [16h, 16h+16)
        v8f acc = {};
#pragma unroll
        for (int kb = 0; kb < 4; ++kb) {
            v2f av, bv;
#pragma unroll
            for (int j = 0; j < 2; ++j) {
                // A 16x4 f32 layout: M = lane&15, K = vgpr + 2*(lane>=16)
                const int ka = kb * 4 + 2 * hi16 + j;
                av[j] = (ka <= lrow) ? 1.0f : 0.0f;        // L[m][k] = (k <= m)
                // B 4x16 f32 layout: N = lane&15, K = vgpr + 2*(lane>=16)
                bv[j] = Pw[ka * C_CH + (lrow + 16 * h)];
            }
            acc = __builtin_amdgcn_wmma_f32_16x16x4_f32(
                false, av, false, bv, (short)0, acc, false, false);
        }
        // ---- Phase 3: scale by exp(-frame_t * rate_c) and store ----
        // C/D 16x16 f32 layout: VGPR i -> M(frame) = i + 8*(lane>=16), N = lane&15
        const int ch = lrow + 16 * h;
        const float rc = softplus_fast(decay[ch]);
#pragma unroll
        for (int i = 0; i < 8; ++i) {
            const int f = i + 8 * hi16;
            const float frame_t = ts + (float)(f + 1) * dt;
            outseg[f * C_CH + ch] = __expf(-frame_t * rc) * acc[i];
        }
    }
}

extern "C" void kernel_launch(void* const* d_in, const int* in_sizes, int n_in,
                              void* d_out, int out_size, void* d_ws, size_t ws_size,
                              hipStream_t stream) {
    const float* features = (const float*)d_in[0];
    const float* times    = (const float*)d_in[1];
    const float* decay    = (const float*)d_in[2];
    const int*   seg_ids  = (const int*)d_in[3];
    // d_in[4] = batch_ids (derived as seg/G on device)
    const float* t_start  = (const float*)d_in[5];
    const float* t_stop   = (const float*)d_in[6];
    // d_in[7] = num_frames (16), d_in[8] = grid_size (512): structural constants of
    // the WMMA tiling; G derived from out_size (device scalars unreadable in capture).

    const int E = in_sizes[1];                 // events
    const int B = in_sizes[5];                 // batch
    const int numSeg = out_size / (F_FR * C_CH);   // B*G segments
    const int G = numSeg / B;

    dim3 block(C_CH * WAVES);
    dim3 grid((numSeg + WAVES - 1) / WAVES);
    GridEmaInterpolate_kernel<<<grid, block, 0, stream>>>(
        features, times, decay, seg_ids, t_start, t_stop,
        (float*)d_out, E, G, numSeg);
}